// EdgeSIMPLEBatched_69183333204158
// MI455X (gfx1250) — compile-verified
//
#include <hip/hip_runtime.h>
#include <stdint.h>

// ---------------------------------------------------------------------------
// SIMPLE exact-k subset sampling (B=8, Nmax=30000, E=4, k=32, T=2) for MI455X.
// Tree DP in log-space; fp32 throughout (matches reference). All scratch fits
// comfortably in the 192MB L2. CDNA5 paths: v_wmma_f32_16x16x4_f32 computes
// the reference's outer-sum S = A (+) B on matrix cores for the 33x33 levels;
// Z-node operands are staged global->LDS via async-to-LDS (ASYNCcnt).
// ---------------------------------------------------------------------------

#define NEGV (-1.0e10f)

typedef float v2f __attribute__((ext_vector_type(2)));
typedef float v8f __attribute__((ext_vector_type(8)));

constexpr int Bc   = 8;
constexpr int Ec   = 4;
constexpr int Nmax = 30000;
constexpr int Rr   = Bc * Ec;      // 32 rows
constexpr int Nn   = 32768;        // padded to 2^15
constexpr int LEV  = 15;
constexpr int KKk  = 32;           // k
constexpr int KC   = 33;           // k+1 counts
constexpr int TT   = 2;

// counts per level: c[0]=2 (leaf {0,score}); c[l]=min(2^l,k)+1
static const int CL_h[16] = {2,3,5,9,17,33,33,33,33,33,33,33,33,33,33,33};

// ------------------------------ RNG (Gumbel) -------------------------------
__device__ __forceinline__ unsigned mix32(unsigned x) {
  x ^= x >> 16; x *= 0x7feb352du;
  x ^= x >> 15; x *= 0x846ca68bu;
  x ^= x >> 16; return x;
}
__device__ __forceinline__ float gumbel_from(unsigned h) {
  float u = ((h >> 8) + 0.5f) * (1.0f / 16777216.0f);   // (0,1)
  return -__logf(-__logf(u));
}

// -------------------- forward conv, small levels (l = 0..4) ----------------
// One thread per (r, out_node, j):  out[j] = logsumexp_a A[a] + B[j-a]
__global__ void k_fwd_small(const float* __restrict__ scores,
                            const float* __restrict__ Zin,
                            float* __restrict__ Zout,
                            int level, int cin, int cout, int Mout)
{
  int gid = blockIdx.x * blockDim.x + threadIdx.x;
  int total = Rr * Mout * cout;
  if (gid >= total) return;
  int j = gid % cout;
  int t = gid / cout;
  int m = t % Mout;
  int r = t / Mout;

  float Av[17], Bv[17];
  if (level == 0) {
    int nA = 2 * m, nB = 2 * m + 1;
    int b = r / Ec, e = r % Ec;
    float sA = (nA < Nmax) ? scores[((size_t)b * Nmax + nA) * Ec + e] : NEGV;
    float sB = (nB < Nmax) ? scores[((size_t)b * Nmax + nB) * Ec + e] : NEGV;
    Av[0] = 0.f; Av[1] = sA; Bv[0] = 0.f; Bv[1] = sB;
  } else {
    const float* A = Zin + ((size_t)r * (2 * Mout) + 2 * m) * cin;
    const float* B = A + cin;
    for (int a = 0; a < cin; ++a) { Av[a] = A[a]; Bv[a] = B[a]; }
  }

  int alo = (j - (cin - 1) > 0) ? j - (cin - 1) : 0;
  int ahi = (cin - 1 < j) ? cin - 1 : j;
  float mx = -3.0e10f;
  for (int a = alo; a <= ahi; ++a) mx = fmaxf(mx, Av[a] + Bv[j - a]);
  float out;
  if (mx <= -1.0e9f) out = NEGV;
  else {
    float s = 0.f;
    for (int a = alo; a <= ahi; ++a) s += __expf(Av[a] + Bv[j - a] - mx);
    out = __logf(s) + mx;
  }
  Zout[((size_t)r * Mout + m) * cout + j] = out;
}

// -------------------- forward conv, 33x33 levels (l = 5..14) ---------------
// One wave (32 threads) per (r, out_node). Async-stage the 66 contiguous
// floats (A|B) into LDS, compute S[a][c] = A[a]+B[c] with chained f32 WMMAs
// (S = A*1^T + 1*B^T), then per-antidiagonal logsumexp.
__global__ void __launch_bounds__(32)
k_fwd_wmma(const float* __restrict__ Zin, float* __restrict__ Zout, int Mout)
{
  __shared__ float sAB[66];        // A[0..32] | B[0..32]
  __shared__ float S[48 * 48];     // padded 3x3 grid of 16x16 tiles

  int m = blockIdx.x % Mout;
  int r = blockIdx.x / Mout;
  int ln = threadIdx.x;
  bool lo = ln < 16;
  int lm = ln & 15;

  const float* gA = Zin + ((size_t)r * (2 * Mout) + 2 * m) * KC;  // 66 floats

  // ---- async global -> LDS staging (ASYNCcnt) ----
  for (int i = ln; i < 66; i += 32) {
    uint32_t ldsa = (uint32_t)(uintptr_t)(&sAB[i]);
    const float* p = gA + i;
    asm volatile("global_load_async_to_lds_b32 %0, %1, off"
                 :: "v"(ldsa), "v"(p) : "memory");
  }
  asm volatile("s_wait_asynccnt 0x0" ::: "memory");
  __syncthreads();

  // ---- outer-sum tiles on matrix cores ----
  for (int ti = 0; ti < 3; ++ti) {
    int ai = 16 * ti + lm;
    float av = (lo && ai < KC) ? sAB[ai] : 0.f;
    for (int tj = 0; tj < 3; ++tj) {
      int bi = 16 * tj + lm;
      float bv = (lo && bi < KC) ? sAB[KC + bi] : 0.f;

      v2f aA;   aA.x = lo ? av : 0.f;   aA.y = 0.f;   // A values on K=0 row
      v2f bOne; bOne.x = lo ? 1.f : 0.f; bOne.y = 0.f; // ones on K=0 row
      v8f acc = {};
      acc = __builtin_amdgcn_wmma_f32_16x16x4_f32(false, aA, false, bOne,
                                                  (short)0, acc, false, false);
      v2f aOne; aOne.x = lo ? 1.f : 0.f; aOne.y = 0.f;
      v2f bB;   bB.x = lo ? bv : 0.f;   bB.y = 0.f;
      acc = __builtin_amdgcn_wmma_f32_16x16x4_f32(false, aOne, false, bB,
                                                  (short)0, acc, false, false);
      // D layout: lane VGPR v holds D[m = v + 8*(lane>=16)][n = lane&15]
      int row0 = 16 * ti + (lo ? 0 : 8);
      int col  = 16 * tj + lm;
#pragma unroll
      for (int v = 0; v < 8; ++v) S[(row0 + v) * 48 + col] = acc[v];
    }
  }
  __syncthreads();

  // ---- per-antidiagonal logsumexp: out[j] = lse_{a+c=j} S[a][c] ----
  for (int j = ln; j < KC; j += 32) {
    int ahi = (j < KC - 1) ? j : KC - 1;
    float mx = -3.0e10f;
    for (int a = 0; a <= ahi; ++a) mx = fmaxf(mx, S[a * 48 + (j - a)]);
    float out;
    if (mx <= -1.0e9f) out = NEGV;
    else {
      float s = 0.f;
      for (int a = 0; a <= ahi; ++a) s += __expf(S[a * 48 + (j - a)] - mx);
      out = __logf(s) + mx;
    }
    Zout[((size_t)r * Mout + m) * KC + j] = out;
  }
}

// ------------------------------- D-pass ------------------------------------
__global__ void k_dinit(float* __restrict__ D)
{
  int gid = blockIdx.x * blockDim.x + threadIdx.x;
  if (gid < Rr * KC) D[gid] = ((gid % KC) == 0) ? 0.f : NEGV;
}

// ref iteration l (14..1): Dout[ch][j] = lse_c Din[ch>>1][j-c] + Z_l[ch^1][c]
__global__ void k_down(const float* __restrict__ Zl,
                       const float* __restrict__ Din,
                       float* __restrict__ Dout, int cl, int Ml)
{
  int gid = blockIdx.x * blockDim.x + threadIdx.x;
  int total = Rr * Ml * KC;
  if (gid >= total) return;
  int j  = gid % KC;
  int t  = gid / KC;
  int ch = t % Ml;
  int r  = t / Ml;
  int p = ch >> 1, sib = ch ^ 1;
  const float* D = Din + ((size_t)r * (Ml / 2) + p) * KC;
  const float* Z = Zl  + ((size_t)r * Ml + sib) * cl;

  int clo = (j - (KC - 1) > 0) ? j - (KC - 1) : 0;
  int chi = (cl - 1 < j) ? cl - 1 : j;
  float mx = -3.0e10f;
  for (int c = clo; c <= chi; ++c) mx = fmaxf(mx, D[j - c] + Z[c]);
  float out;
  if (mx <= -1.0e9f) out = NEGV;
  else {
    float s = 0.f;
    for (int c = clo; c <= chi; ++c) {
      float dv = D[j - c];
      if (dv > -1.0e9f) s += __expf(dv + Z[c] - mx);
    }
    out = (s > 0.f) ? __logf(s) + mx : NEGV;
  }
  Dout[((size_t)r * Ml + ch) * KC + j] = out;
}

// ------------------- marginals (fuses leaf-level D step) -------------------
__global__ void k_marg(const float* __restrict__ scores,
                       const float* __restrict__ D1,   // [R][16384][33]
                       const float* __restrict__ Z15,  // [R][1][33]
                       float* __restrict__ out_marg)
{
  int gid = blockIdx.x * blockDim.x + threadIdx.x;
  if (gid >= Rr * Nmax) return;
  int n = gid % Nmax;
  int r = gid / Nmax;
  int b = r / Ec, e = r % Ec;

  float sc = scores[((size_t)b * Nmax + n) * Ec + e];
  int p = n >> 1, sib = n ^ 1;
  float sibsc = (sib < Nmax) ? scores[((size_t)b * Nmax + sib) * Ec + e] : NEGV;
  const float* D = D1 + ((size_t)r * (Nn >> 1) + p) * KC;

  // D_leaf[n][k-1] = logaddexp(D[p][k-1] + 0, D[p][k-2] + score_sib)
  float x0 = D[KKk - 1];
  float x1 = D[KKk - 2] + sibsc;
  float mx = fmaxf(x0, x1), dl;
  if (mx <= -1.0e9f) dl = NEGV;
  else dl = __logf(__expf(x0 - mx) + __expf(x1 - mx)) + mx;

  float logZk = Z15[(size_t)r * KC + KKk];
  float mg = __expf(sc + dl - logZk);
  out_marg[((size_t)b * Nmax + n) * Ec + e] = mg;
}

// ------------------------------ sampling -----------------------------------
__global__ void k_qinit(int* __restrict__ q)
{
  int gid = blockIdx.x * blockDim.x + threadIdx.x;
  if (gid < Rr) q[gid] = KKk;
}

// ref iteration l (14..1): Gumbel-argmax over a of ZL[a] + ZR[q-a]
__global__ void k_sample(const float* __restrict__ Zl,
                         const int* __restrict__ qin, int* __restrict__ qout,
                         int cl, int Min, int level, int t, unsigned seed)
{
  int gid = blockIdx.x * blockDim.x + threadIdx.x;
  if (gid >= Rr * Min) return;
  int m = gid % Min;
  int r = gid / Min;
  int qv = qin[(size_t)r * Min + m];
  const float* ZL = Zl + ((size_t)r * (2 * Min) + 2 * m) * cl;
  const float* ZR = ZL + cl;

  float best = -3.4e38f; int bj = 0;
  for (int a = 0; a < cl; ++a) {
    int c = qv - a;
    float lg = (c >= 0 && c < cl) ? ZL[a] + ZR[c] : NEGV;
    unsigned h = mix32(((((seed ^ (unsigned)t) * 0x01000193u ^ (unsigned)level)
                        * 0x01000193u ^ (unsigned)r) * 0x01000193u ^ (unsigned)m)
                        * 0x01000193u ^ (unsigned)a);
    float v = lg + gumbel_from(h);
    if (v > best) { best = v; bj = a; }
  }
  qout[(size_t)r * (2 * Min) + 2 * m]     = bj;
  qout[(size_t)r * (2 * Min) + 2 * m + 1] = qv - bj;
}

// fused leaf-level sampling + straight-through mask write
__global__ void k_mask(const float* __restrict__ scores,
                       const int* __restrict__ qin,     // [R][16384]
                       const float* __restrict__ marg,  // [B][Nmax][E]
                       float* __restrict__ out_mask,    // [B][Nmax][E] (t slice)
                       int t, unsigned seed)
{
  int gid = blockIdx.x * blockDim.x + threadIdx.x;
  const int Min = Nn >> 1;
  if (gid >= Rr * Min) return;
  int m = gid % Min;
  int r = gid / Min;
  int b = r / Ec, e = r % Ec;
  int qv = qin[(size_t)r * Min + m];
  int nA = 2 * m, nB = 2 * m + 1;
  float sA = (nA < Nmax) ? scores[((size_t)b * Nmax + nA) * Ec + e] : NEGV;
  float sB = (nB < Nmax) ? scores[((size_t)b * Nmax + nB) * Ec + e] : NEGV;

  float best = -3.4e38f; int bj = 0;
  for (int a = 0; a < 2; ++a) {
    int c = qv - a;
    float lg = (c >= 0 && c < 2) ? (a ? sA : 0.f) + (c ? sB : 0.f) : NEGV;
    unsigned h = mix32(((((seed ^ (unsigned)t) * 0x01000193u)
                        * 0x01000193u ^ (unsigned)r) * 0x01000193u ^ (unsigned)m)
                        * 0x01000193u ^ (unsigned)a);
    float v = lg + gumbel_from(h);
    if (v > best) { best = v; bj = a; }
  }
  int qL = bj, qR = qv - bj;
  if (nA < Nmax) {
    float mg = marg[((size_t)b * Nmax + nA) * Ec + e];
    out_mask[((size_t)b * Nmax + nA) * Ec + e] = ((float)qL - mg) + mg;
  }
  if (nB < Nmax) {
    float mg = marg[((size_t)b * Nmax + nB) * Ec + e];
    out_mask[((size_t)b * Nmax + nB) * Ec + e] = ((float)qR - mg) + mg;
  }
}

// ------------------------------- launcher ----------------------------------
extern "C" void kernel_launch(void* const* d_in, const int* in_sizes, int n_in,
                              void* d_out, int out_size, void* d_ws, size_t ws_size,
                              hipStream_t stream)
{
  (void)in_sizes; (void)n_in; (void)out_size; (void)ws_size;
  const float* scores = (const float*)d_in[0];
  float* out      = (float*)d_out;
  float* out_mask = out;                                   // [T,B,Nmax,E]
  float* out_marg = out + (size_t)TT * Bc * Nmax * Ec;     // [B,Nmax,E]

  // workspace layout (floats)
  float* W = (float*)d_ws;
  size_t zoff[16]; size_t cur = 0;
  for (int l = 1; l <= LEV; ++l) { zoff[l] = cur; cur += (size_t)Rr * (size_t)(Nn >> l) * CL_h[l]; }
  float* Y = W + cur;                            // D ping (cap 16384 nodes)
  float* X = Y + (size_t)Rr * (Nn >> 1) * KC;    // D pong (cap 8192 nodes)
  int*   qY = (int*)(X + (size_t)Rr * (Nn >> 2) * KC);
  int*   qX = qY + (size_t)Rr * (Nn >> 1);

  auto blocks = [](size_t total) { return dim3((unsigned)((total + 255) / 256)); };

  // ---- forward tree build ----
  for (int l = 0; l <= 4; ++l) {
    int Mout = Nn >> (l + 1);
    size_t total = (size_t)Rr * Mout * CL_h[l + 1];
    const float* Zin = (l == 0) ? nullptr : (W + zoff[l]);
    k_fwd_small<<<blocks(total), 256, 0, stream>>>(scores, Zin, W + zoff[l + 1],
                                                   l, CL_h[l], CL_h[l + 1], Mout);
  }
  for (int l = 5; l <= 14; ++l) {
    int Mout = Nn >> (l + 1);
    k_fwd_wmma<<<dim3(Rr * Mout), 32, 0, stream>>>(W + zoff[l], W + zoff[l + 1], Mout);
  }

  // ---- downsweep (D) ----
  k_dinit<<<blocks((size_t)Rr * KC), 256, 0, stream>>>(Y);
  for (int l = 14; l >= 1; --l) {
    int Ml = Nn >> l;                       // child count at this iteration
    float* Din  = ((l + 1) % 2 == 0) ? X : Y;
    float* Dout = (l % 2 == 0) ? X : Y;
    size_t total = (size_t)Rr * Ml * KC;
    k_down<<<blocks(total), 256, 0, stream>>>(W + zoff[l], Din, Dout, CL_h[l], Ml);
  }
  // final D (16384 nodes) is in Y
  k_marg<<<blocks((size_t)Rr * Nmax), 256, 0, stream>>>(scores, Y, W + zoff[15], out_marg);

  // ---- sampling (T descents) ----
  for (int t = 0; t < TT; ++t) {
    k_qinit<<<1, 64, 0, stream>>>(qY);
    for (int l = 14; l >= 1; --l) {
      int Min = Nn >> (l + 1);
      int* qin  = ((l + 1) % 2 == 0) ? qX : qY;
      int* qout = (l % 2 == 0) ? qX : qY;
      size_t total = (size_t)Rr * Min;
      k_sample<<<blocks(total), 256, 0, stream>>>(W + zoff[l], qin, qout,
                                                  CL_h[l], Min, l, t, 0x12345679u);
    }
    size_t total = (size_t)Rr * (Nn >> 1);
    k_mask<<<blocks(total), 256, 0, stream>>>(scores, qY, out_marg,
                                              out_mask + (size_t)t * Bc * Nmax * Ec,
                                              t, 0x12345679u);
  }
}